// ApplyPolicyMap_1297080123727
// MI455X (gfx1250) — compile-verified
//
#include <hip/hip_runtime.h>
#include <stdint.h>

// ApplyPolicyMap: out[b,m] = inputs_flat[b, row[m]]  (pmap is one-hot per column,
// so the reference matmul is an exact fp32 gather — 156 GFLOP of dense WMMA work
// collapses to ~267 MB of data movement; memory-roofline ~11.5 us at 23.3 TB/s).
// Strategy: (1) pmap -> row[] indices, (2) TDM-DMA input rows to LDS, gather.

#define C_IN    5120
#define N_MOVES 1858
#define NB      2        // batch rows per workgroup (2*5120*4B = 40 KB LDS tile)
#define THREADS 256

typedef __attribute__((ext_vector_type(4))) unsigned int v4u;
typedef __attribute__((ext_vector_type(8))) int          v8i;
typedef __attribute__((ext_vector_type(4))) int          v4i;

// ---------------------------------------------------------------------------
// Kernel 1: pmap [5120,1858] one-hot columns -> row index per column.
// Coalesced across m; pmap is read once and never again -> non-temporal loads.
// Exactly one thread writes idx[m] per call (each column has exactly one 1.0),
// so d_ws contents are fully deterministic every launch.
// ---------------------------------------------------------------------------
__global__ void build_index_kernel(const float* __restrict__ pmap,
                                   int* __restrict__ idx) {
  int m = blockIdx.x * blockDim.x + threadIdx.x;
  if (m >= N_MOVES) return;
  int r0 = blockIdx.y * 256;                 // 5120/256 = 20 row chunks
  const float* p = pmap + (size_t)r0 * N_MOVES + m;
  for (int r = 0; r < 256; ++r) {
    float v = __builtin_nontemporal_load(p + (size_t)r * N_MOVES);
    if (v != 0.0f) idx[m] = r0 + r;
  }
}

// ---------------------------------------------------------------------------
// Kernel 2: per block, TDM-load NB batch rows (2-D tile 5120 x NB, f32) into
// LDS while all threads stage idx[] into LDS, then gather and store coalesced
// + non-temporal (output is write-once; keep L2 for the input stream).
// ---------------------------------------------------------------------------
__global__ __launch_bounds__(THREADS)
void policy_gather_kernel(const float* __restrict__ in,
                          const int* __restrict__ idx,
                          float* __restrict__ out, int B) {
  __shared__ float smem[NB * C_IN];          // 40 KB input tile
  __shared__ int   sidx[N_MOVES];            // 7.3 KB index table

  const int b0  = blockIdx.x * NB;
  int rem = B - b0; if (rem > NB) rem = NB;  // B=8192 -> always NB

  if (threadIdx.x < 32) {                    // wave 0 drives the Tensor Data Mover
    // Low 32 bits of a generic LDS pointer == LDS byte offset (ISA 10.2).
    unsigned lds_off = (unsigned)(uintptr_t)&smem[0];
    unsigned long long ga =
        (unsigned long long)(uintptr_t)(in + (size_t)b0 * C_IN);

    // D# group 0: count=1 | lds_addr | global_addr[56:0] | type=2
    v4u g0;
    g0.x = 1u;
    g0.y = lds_off;
    g0.z = (unsigned)(ga & 0xFFFFFFFFu);
    g0.w = (unsigned)((ga >> 32) & 0x01FFFFFFu) | 0x80000000u;  // type=2 -> [127:126]

    // D# group 1: data_size=4B; tensor 5120 x rem; tile 5120 x rem; stride0=5120
    v8i g1;
    g1[0] = (2 << 16);                               // data_size=2 (4 bytes), mask=0
    g1[1] = (int)((C_IN & 0xFFFF) << 16);            // tensor_dim0[15:0] -> bits[63:48]
    g1[2] = (C_IN >> 16) | (rem << 16);              // tensor_dim0 hi | tensor_dim1 lo
    g1[3] = (int)((unsigned)(C_IN & 0xFFFF) << 16);  // tensor_dim1 hi(0) | tile_dim0
    g1[4] = rem;                                     // tile_dim1 | tile_dim2=0
    g1[5] = C_IN;                                    // tensor_dim0_stride[31:0]
    g1[6] = 0;                                       // stride0 hi | stride1 lo
    g1[7] = 0;                                       // stride1 hi

    v4i gz = {0, 0, 0, 0};                           // groups 2/3: <=2-D tensor
#if __clang_major__ >= 23
    v8i gz8 = {0, 0, 0, 0, 0, 0, 0, 0};
    __builtin_amdgcn_tensor_load_to_lds(g0, g1, gz, gz, gz8, 0);
#else
    __builtin_amdgcn_tensor_load_to_lds(g0, g1, gz, gz, 0);
#endif
  }

  // Overlap with the DMA: stage the (hot-in-L2) index table into LDS.
  for (int m = threadIdx.x; m < N_MOVES; m += THREADS) sidx[m] = idx[m];

  if (threadIdx.x < 32) __builtin_amdgcn_s_wait_tensorcnt(0);
  __syncthreads();                                   // publish tile + indices

  const int total = rem * N_MOVES;
  for (int i = threadIdx.x; i < total; i += THREADS) {
    int bi = i / N_MOVES;
    int m  = i - bi * N_MOVES;
    float v = smem[bi * C_IN + sidx[m]];             // on-chip random access (ds_load)
    __builtin_nontemporal_store(v, out + (size_t)(b0 + bi) * N_MOVES + m);
  }
}

// ---------------------------------------------------------------------------
extern "C" void kernel_launch(void* const* d_in, const int* in_sizes, int n_in,
                              void* d_out, int out_size, void* d_ws, size_t ws_size,
                              hipStream_t stream) {
  const float* inputs = (const float*)d_in[0];   // [B, 80, 8, 8] f32
  const float* pmap   = (const float*)d_in[1];   // [5120, 1858] f32 one-hot cols
  float*       out    = (float*)d_out;           // [B, 1858] f32
  int*         idx    = (int*)d_ws;              // 1858 ints scratch

  const int B = in_sizes[0] / C_IN;              // 8192

  dim3 gIdx((N_MOVES + 255) / 256, C_IN / 256);  // (8, 20)
  build_index_kernel<<<gIdx, 256, 0, stream>>>(pmap, idx);

  int nblk = (B + NB - 1) / NB;                  // 4096 workgroups
  policy_gather_kernel<<<nblk, THREADS, 0, stream>>>(inputs, idx, out, B);
}